// CustomGCN_21268678050245
// MI455X (gfx1250) — compile-verified
//
#include <hip/hip_runtime.h>
#include <hip/hip_bf16.h>

typedef __attribute__((ext_vector_type(2))) float v2f;
typedef __attribute__((ext_vector_type(8))) float v8f;

#define GD 128          // feature dim D
#define BN_EPS 1e-5f
#define LRELU_SLOPE 0.01f

// ---------------------------------------------------------------------------
// WMMA fp32 GEMM: C[n x 128] = A[n x 128] @ W[128 x 128]   (W row-major)
// One block = 256 threads = 8 waves; block computes 128 rows x 128 cols.
// Each wave: 16-row stripe, 8 accumulators of v8f (16x16 tiles across N).
// W staged in LDS (64KB). Uses V_WMMA_F32_16X16X4_F32, K-loop of 32 steps.
// A-frag layout (ISA 7.12.2, 32-bit A 16x4): lanes0-15 K=k0,k0+1 ; lanes16-31 K=k0+2,k0+3
// B-frag: lanes0-15 rows K=k0..k0+1 (V0,V1); lanes16-31 rows K=k0+2..k0+3
// C/D: lane = n + 16*(m>=8), vgpr r = m%8
// ---------------------------------------------------------------------------
__global__ __launch_bounds__(256)
void wmma_gemm128(const float* __restrict__ A, const float* __restrict__ W,
                  float* __restrict__ C, int nrows) {
    __shared__ float lds[GD * GD];  // 64KB: W row-major
    const int t = threadIdx.x;
    for (int i = t; i < GD * GD; i += 256) lds[i] = W[i];
    __syncthreads();

    const int wave = t >> 5;
    const int lane = t & 31;
    const int half = lane >> 4;   // 0: lanes 0-15, 1: lanes 16-31
    const int l16  = lane & 15;

    const int m0 = blockIdx.x * 128 + wave * 16;
    int arow = m0 + l16;
    if (arow >= nrows) arow = nrows - 1;          // clamp loads; stores guarded
    const float* aptr = A + (size_t)arow * GD + half * 2;

    v8f acc[8];
#pragma unroll
    for (int nt = 0; nt < 8; ++nt) acc[nt] = (v8f){0, 0, 0, 0, 0, 0, 0, 0};

    for (int k0 = 0; k0 < GD; k0 += 4) {
        const v2f a = *(const v2f*)(aptr + k0);   // 8B-aligned global_load_b64
#pragma unroll
        for (int nt = 0; nt < 8; ++nt) {
            const int nn = nt * 16 + l16;
            v2f b;
            b.x = lds[(k0 + half * 2)     * GD + nn];
            b.y = lds[(k0 + half * 2 + 1) * GD + nn];
            acc[nt] = __builtin_amdgcn_wmma_f32_16x16x4_f32(
                false, a, false, b, (short)0, acc[nt], false, false);
        }
    }

#pragma unroll
    for (int r = 0; r < 8; ++r) {
        const int row = m0 + r + half * 8;
        if (row < nrows) {
#pragma unroll
            for (int nt = 0; nt < 8; ++nt)
                C[(size_t)row * GD + nt * 16 + l16] = acc[nt][r];
        }
    }
}

// ---------------------------------------------------------------------------
// degree / dinv precompute
// ---------------------------------------------------------------------------
__global__ void init_one_kernel(float* __restrict__ p, int n) {
    int i = blockIdx.x * 256 + threadIdx.x;
    if (i < n) p[i] = 1.0f;
}
__global__ void deg_accum_kernel(float* __restrict__ deg, const int* __restrict__ dst, int e) {
    int i = blockIdx.x * 256 + threadIdx.x;
    if (i < e) unsafeAtomicAdd(&deg[dst[i]], 1.0f);
}
__global__ void rsqrt_kernel(const float* __restrict__ deg, float* __restrict__ dinv, int n) {
    int i = blockIdx.x * 256 + threadIdx.x;
    if (i < n) dinv[i] = rsqrtf(deg[i]);
}
__global__ void copy_kernel(const float* __restrict__ a, float* __restrict__ b, int n) {
    int i = blockIdx.x * 256 + threadIdx.x;
    if (i < n) b[i] = a[i];
}
__global__ void zero_kernel(float* __restrict__ p, int n) {
    int i = blockIdx.x * 256 + threadIdx.x;
    if (i < n) p[i] = 0.0f;
}

// ---------------------------------------------------------------------------
// edge scatter: one wave32 per edge, float4 per lane (128 floats/edge),
// global_atomic_add_f32 into agg[dst]
// ---------------------------------------------------------------------------
__global__ __launch_bounds__(256)
void scatter_kernel(const float* __restrict__ h, const int* __restrict__ src,
                    const int* __restrict__ dst, const float* __restrict__ dinv,
                    float* __restrict__ agg, int e) {
    int gid = blockIdx.x * 256 + threadIdx.x;
    int edge = gid >> 5;
    if (edge >= e) return;
    int lane = gid & 31;
    int s = src[edge], d = dst[edge];
    float w = dinv[s] * dinv[d];
    const float4 v = *((const float4*)(h + (size_t)s * GD) + lane);
    float* ad = agg + (size_t)d * GD + lane * 4;
    unsafeAtomicAdd(ad + 0, v.x * w);
    unsafeAtomicAdd(ad + 1, v.y * w);
    unsafeAtomicAdd(ad + 2, v.z * w);
    unsafeAtomicAdd(ad + 3, v.w * w);
}

// ---------------------------------------------------------------------------
// elementwise epilogues (x_skip == x invariant lets these run in place)
// ---------------------------------------------------------------------------
__global__ void post_conv_kernel(float* __restrict__ x, const float* __restrict__ h,
                                 const float* __restrict__ agg, const float* __restrict__ dinv,
                                 const float* __restrict__ bc, int nd) {
    int i = blockIdx.x * 256 + threadIdx.x;
    if (i >= nd) return;
    int row = i >> 7, col = i & (GD - 1);
    float di = dinv[row];
    float conv = agg[i] + h[i] * di * di + bc[col];
    float v = conv + x[i];               // skip-add (x_skip == old x)
    x[i] = v > 0.0f ? v : 0.0f;          // relu
}
__global__ void post_ff_kernel(float* __restrict__ x, const float* __restrict__ t,
                               const float* __restrict__ bf, int nd) {
    int i = blockIdx.x * 256 + threadIdx.x;
    if (i >= nd) return;
    int col = i & (GD - 1);
    float u = t[i] + bf[col];
    u = u > 0.0f ? u : LRELU_SLOPE * u;  // leaky_relu
    float v = u + x[i];                  // skip-add
    x[i] = v > 0.0f ? v : 0.0f;          // relu
}
__global__ void post_skip_kernel(float* __restrict__ x, const float* __restrict__ t,
                                 const float* __restrict__ bskip, int nd) {
    int i = blockIdx.x * 256 + threadIdx.x;
    if (i >= nd) return;
    int col = i & (GD - 1);
    float s = t[i] + bskip[col];         // transformed skip branch
    float v = x[i] + s;
    x[i] = v > 0.0f ? v : 0.0f;          // relu
}
__global__ void bn_kernel(float* __restrict__ x, const float* __restrict__ gamma,
                          const float* __restrict__ beta, const float* __restrict__ rmean,
                          const float* __restrict__ rvar, int nd) {
    int i = blockIdx.x * 256 + threadIdx.x;
    if (i >= nd) return;
    int col = i & (GD - 1);
    float xv = x[i];
    float b = (xv - rmean[col]) * rsqrtf(rvar[col] + BN_EPS) * gamma[col] + beta[col];
    float v = b + xv;                    // skip-add (x_skip == x)
    x[i] = v > 0.0f ? v : 0.0f;          // relu
}

// ---------------------------------------------------------------------------
extern "C" void kernel_launch(void* const* d_in, const int* in_sizes, int n_in,
                              void* d_out, int out_size, void* d_ws, size_t ws_size,
                              hipStream_t stream) {
    const float* x_in  = (const float*)d_in[0];
    const int*   ei    = (const int*)d_in[1];
    const float* Wc    = (const float*)d_in[2];
    const float* bc    = (const float*)d_in[3];
    const float* Wf    = (const float*)d_in[4];
    const float* bf    = (const float*)d_in[5];
    const float* Wskip = (const float*)d_in[6];
    const float* bskip = (const float*)d_in[7];
    const float* gamma = (const float*)d_in[8];
    const float* beta  = (const float*)d_in[9];
    const float* rmean = (const float*)d_in[10];
    const float* rvar  = (const float*)d_in[11];

    const int n = in_sizes[0] / GD;       // 50000 nodes
    const int e = in_sizes[1] / 2;        // 500000 edges
    const int L = in_sizes[3] / GD;       // 3 layers (bc is [L, D])

    const int* src = ei;                  // edge_index[0]
    const int* dst = ei + e;              // edge_index[1]

    float* x    = (float*)d_out;          // persistent node features (x == x_skip)
    float* bufA = (float*)d_ws;           // GEMM outputs (h / t / skip)
    float* bufB = bufA + (size_t)n * GD;  // aggregation buffer
    float* deg  = bufB + (size_t)n * GD;  // [n]
    float* dinv = deg + n;                // [n]

    const int ND = n * GD;
    dim3 blk(256);
    const int gND   = (ND + 255) / 256;
    const int gN    = (n + 255) / 256;
    const int gE    = (e + 255) / 256;
    const int gE32  = (int)(((long long)e * 32 + 255) / 256);
    const int gGemm = (n + 127) / 128;

    // degree with self loops; dinv = deg^-1/2
    init_one_kernel<<<gN, blk, 0, stream>>>(deg, n);
    deg_accum_kernel<<<gE, blk, 0, stream>>>(deg, dst, e);
    rsqrt_kernel<<<gN, blk, 0, stream>>>(deg, dinv, n);
    copy_kernel<<<gND, blk, 0, stream>>>(x_in, x, ND);

    for (int i = 0; i < L; ++i) {
        // GCNConv: h = x @ Wc[i]; agg = scatter(norm * h[src] -> dst); self loop + bias
        wmma_gemm128<<<gGemm, blk, 0, stream>>>(x, Wc + (size_t)i * GD * GD, bufA, n);
        zero_kernel<<<gND, blk, 0, stream>>>(bufB, ND);
        scatter_kernel<<<gE32, blk, 0, stream>>>(bufA, src, dst, dinv, bufB, e);
        post_conv_kernel<<<gND, blk, 0, stream>>>(x, bufA, bufB, dinv, bc + i * GD, ND);

        // feed-forward: leaky_relu(x@Wf + bf), skip+relu
        wmma_gemm128<<<gGemm, blk, 0, stream>>>(x, Wf + (size_t)i * GD * GD, bufA, n);
        post_ff_kernel<<<gND, blk, 0, stream>>>(x, bufA, bf + i * GD, ND);

        if (i > 0) {
            // skip transform: x_skip(@==x) @ Wskip + bskip, skip+relu
            wmma_gemm128<<<gGemm, blk, 0, stream>>>(x, Wskip + (size_t)(i - 1) * GD * GD, bufA, n);
            post_skip_kernel<<<gND, blk, 0, stream>>>(x, bufA, bskip + (i - 1) * GD, ND);
        }

        // BatchNorm (eval, running stats), skip+relu
        bn_kernel<<<gND, blk, 0, stream>>>(x, gamma + i * GD, beta + i * GD,
                                           rmean + i * GD, rvar + i * GD, ND);
    }
}